// TemporalGNN_18811956756717
// MI455X (gfx1250) — compile-verified
//
#include <hip/hip_runtime.h>
#include <hip/hip_bf16.h>
#include <math.h>
#include <stddef.h>

typedef float v2f __attribute__((ext_vector_type(2)));
typedef float v8f __attribute__((ext_vector_type(8)));

#define D_IN    64
#define H_DIM   128
#define N_ITEMS 2048

// -------------------------------------------------- async global->LDS (CDNA5 path)
#if defined(__has_builtin)
#  if __has_builtin(__builtin_amdgcn_global_load_async_to_lds_b128)
#    define HAVE_ASYNC_B128 1
#  endif
#  if __has_builtin(__builtin_amdgcn_s_wait_asynccnt)
#    define HAVE_WAIT_ASYNC 1
#  endif
#endif

typedef int v4i_gnu __attribute__((vector_size(16)));          // matches builtin param
typedef __attribute__((address_space(1))) v4i_gnu as1_v4i;     // global (AS1)
typedef __attribute__((address_space(3))) v4i_gnu as3_v4i;     // LDS (AS3)

// copy 16 bytes global -> LDS; async (bypasses VGPRs, ASYNCcnt) when available
__device__ __forceinline__ void lds_copy16(float* ldst, const float* gsrc) {
#ifdef HAVE_ASYNC_B128
    as1_v4i* g = (as1_v4i*)(unsigned long long)(size_t)gsrc;   // flat addr == AS1 addr
    as3_v4i* l = (as3_v4i*)(unsigned int)(size_t)ldst;         // low 32 bits == LDS offset
    __builtin_amdgcn_global_load_async_to_lds_b128(g, l, 0, 0);
#else
    *(float4*)ldst = *(const float4*)gsrc;
#endif
}

__device__ __forceinline__ void lds_copy_wait() {
#ifdef HAVE_ASYNC_B128
#  ifdef HAVE_WAIT_ASYNC
    __builtin_amdgcn_s_wait_asynccnt(0);
#  else
    asm volatile("s_wait_asynccnt 0x0" ::: "memory");
#  endif
#endif
}

// ---------------------------------------------------------------- utilities
__global__ void zero_b128_kernel(uint4* __restrict__ p, long long n4) {
    long long i = (long long)blockIdx.x * blockDim.x + threadIdx.x;
    if (i < n4) p[i] = make_uint4(0u, 0u, 0u, 0u);
}

__global__ void degree_kernel(const int* __restrict__ dst, float* __restrict__ deg,
                              int E, int n) {
    int i = blockIdx.x * blockDim.x + threadIdx.x;
    if (i < E) {
        atomicAdd(&deg[dst[i]], 1.0f);
    } else if (i < E + n) {
        atomicAdd(&deg[i - E], 1.0f);   // self-loop
    }
}

__global__ void dinv_kernel(const float* __restrict__ deg, float* __restrict__ dinv, int n) {
    int i = blockIdx.x * blockDim.x + threadIdx.x;
    if (i < n) {
        float d = deg[i];
        dinv[i] = (d > 0.0f) ? rsqrtf(d) : 0.0f;
    }
}

// ------------------------------------------------- GEMM: Y[nrows x 128] = X[nrows x K] @ W[K x 128]
// Block = 128 threads (4 waves) -> 64 rows. X tile staged into LDS with async
// b128 global->LDS loads (coalesced); A fragments come from LDS (+4 dword pad,
// bank-conflict-free). WMMA f32 16x16x4; each wave owns a 16x128 output strip.
template<int K>
__global__ void gemm_wmma_kernel(const float* __restrict__ X, const float* __restrict__ W,
                                 float* __restrict__ Y, int nrows) {
    constexpr int LDK = K + 4;                       // pad: bank stride 4 -> conflict-free
    __shared__ float xt[64 * LDK];

    const int block_row0 = blockIdx.x * 64;
    if (block_row0 >= nrows) return;                 // block-uniform (barrier-safe)

    const int tid = threadIdx.x;

    // cooperative coalesced staging of 64 x K tile (async global->LDS)
#pragma unroll
    for (int it = 0; it < K / 8; ++it) {
        const int flat = (it * 128 + tid) * 4;       // element index in 64*K
        const int r = flat / K;
        const int c = flat & (K - 1);
        lds_copy16(&xt[r * LDK + c], X + (long long)(block_row0 + r) * K + c);
    }
    lds_copy_wait();
    __syncthreads();

    const int lane  = threadIdx.x & 31;
    const int lrow  = lane & 15;                     // A row / B col within tile
    const int khalf = (lane >> 4) << 1;              // 0 or 2
    const int rl    = ((tid >> 5) << 4) + lrow;      // this lane's local A row
    const int row0  = block_row0 + ((tid >> 5) << 4);

    v8f vzero = {};
    v8f acc[8];
#pragma unroll
    for (int t = 0; t < 8; ++t) acc[t] = vzero;

    for (int k0 = 0; k0 < K; k0 += 4) {
        const int ka = k0 + khalf;
        const v2f a = *(const v2f*)&xt[rl * LDK + ka];
#pragma unroll
        for (int t = 0; t < 8; ++t) {
            const int col = t * 16 + lrow;
            v2f b;
            b.x = W[ka * H_DIM + col];
            b.y = W[(ka + 1) * H_DIM + col];
            acc[t] = __builtin_amdgcn_wmma_f32_16x16x4_f32(
                false, a, false, b, (short)0, acc[t], false, false);
        }
    }

    // C/D layout: acc[t][j] -> row = row0 + j + (lane>=16 ? 8:0), col = t*16 + lrow
    const int rb = row0 + ((lane >> 4) << 3);
#pragma unroll
    for (int t = 0; t < 8; ++t) {
        const int col = t * 16 + lrow;
#pragma unroll
        for (int j = 0; j < 8; ++j) {
            Y[(long long)(rb + j) * H_DIM + col] = acc[t][j];
        }
    }
}

// ---------------------------------------------------- scatter: out[dst] += h[src] * dinv[src]*dinv[dst]
// One 32-lane group per edge; float4 row traffic, f32 atomics (reference semantics).
__global__ void scatter_kernel(const float* __restrict__ H, const int* __restrict__ src,
                               const int* __restrict__ dst, const float* __restrict__ dinv,
                               float* __restrict__ out, int E, int n) {
    const int g    = blockIdx.x * (blockDim.x >> 5) + (threadIdx.x >> 5);
    const int lane = threadIdx.x & 31;
    if (g >= E + n) return;                          // uniform per 32-lane group
    int s, d;
    if (g < E) { s = src[g]; d = dst[g]; }
    else       { s = d = g - E; }                    // self-loop
    const float norm = dinv[s] * dinv[d];
    const float4 h4 = *(const float4*)(H + (long long)s * H_DIM + lane * 4);
    float* o = out + (long long)d * H_DIM + lane * 4;
    atomicAdd(o + 0, h4.x * norm);
    atomicAdd(o + 1, h4.y * norm);
    atomicAdd(o + 2, h4.z * norm);
    atomicAdd(o + 3, h4.w * norm);
}

template<bool RELU>
__global__ void bias_act_kernel(const float4* __restrict__ in, const float4* __restrict__ bias,
                                float4* __restrict__ out, long long total4) {
    long long i = (long long)blockIdx.x * blockDim.x + threadIdx.x;
    if (i < total4) {
        const float4 v = in[i];
        const float4 b = bias[i & (H_DIM / 4 - 1)];
        float4 r;
        r.x = v.x + b.x; r.y = v.y + b.y; r.z = v.z + b.z; r.w = v.w + b.w;
        if (RELU) {
            r.x = fmaxf(r.x, 0.0f); r.y = fmaxf(r.y, 0.0f);
            r.z = fmaxf(r.z, 0.0f); r.w = fmaxf(r.w, 0.0f);
        }
        out[i] = r;
    }
}

// ------------------------------------------------------------- unique item count -> num_users
__global__ void item_flag_kernel(const int* __restrict__ items, int* __restrict__ flags, int B) {
    int i = blockIdx.x * blockDim.x + threadIdx.x;
    if (i < B) atomicOr(&flags[items[i]], 1);
}

__global__ void count_unique_kernel(const int* __restrict__ flags, int* __restrict__ numUsers,
                                    int n, int nitems) {
    __shared__ int ssum[256];
    int t = threadIdx.x;
    int c = 0;
    for (int i = t; i < nitems; i += 256) c += (flags[i] != 0);
    ssum[t] = c;
    __syncthreads();
    for (int s = 128; s > 0; s >>= 1) {
        if (t < s) ssum[t] += ssum[t + s];
        __syncthreads();
    }
    if (t == 0) *numUsers = n - ssum[0];
}

// ------------------------------------------------------------------------- MLP scorer
// One wave -> 16 batch rows. Gathered user/item rows staged into LDS with async
// b128 loads; fc1 (256->64) and fc2 (64->32) via WMMA f32 16x16x4; z1/z2 alias
// the zbuf LDS region after barriers. Final 32->1 dot + sigmoid.
#define ZB_LD  260                                   // 256 + 4 pad (bank stride 4)
#define Z1_LD  68                                    // 64 + 4 pad
#define Z2_LD  36                                    // 32 + 4 pad
#define WAVE_LDS (16 * ZB_LD)                        // z1/z2 alias inside

__global__ void mlp_wmma_kernel(const float* __restrict__ h2,
                                const int* __restrict__ users, const int* __restrict__ items,
                                const int* __restrict__ numUsersPtr,
                                const float* __restrict__ fc1_w, const float* __restrict__ fc1_b,
                                const float* __restrict__ fc2_w, const float* __restrict__ fc2_b,
                                const float* __restrict__ out_w, const float* __restrict__ out_b,
                                float* __restrict__ out, int B) {
    __shared__ float lds[4 * WAVE_LDS];

    const int row0_blk = blockIdx.x * 64;
    if (row0_blk >= B) return;                       // block-uniform (barrier-safe)

    const int wib  = threadIdx.x >> 5;               // wave in block (0..3)
    const int lane = threadIdx.x & 31;
    const int row0 = row0_blk + wib * 16;

    const int lrow  = lane & 15;
    const int khalf = (lane >> 4) << 1;
    const int rb    = (lane >> 4) << 3;              // 0 or 8 (C-layout row base)
    const int nu    = *numUsersPtr;

    float* zbuf = lds + wib * WAVE_LDS;              // [16][ZB_LD] : [user(128) | item(128)]
    float* z1   = zbuf;                              // [16][Z1_LD] (aliased after barrier)
    float* z2   = zbuf + 16 * Z1_LD;                 // [16][Z2_LD]

    // ---- stage 16 gathered rows (user||item); coalesced async b128 copies ----
    for (int r = 0; r < 16; ++r) {
        const long long urow = (long long)users[row0 + r] * H_DIM;
        const long long irow = (long long)(items[row0 + r] + nu) * H_DIM;
        lds_copy16(&zbuf[r * ZB_LD + lane * 4],       h2 + urow + lane * 4);
        lds_copy16(&zbuf[r * ZB_LD + 128 + lane * 4], h2 + irow + lane * 4);
    }
    lds_copy_wait();
    __syncthreads();

    v8f vzero = {};

    // ---- fc1: [16 x 256] @ [256 x 64] ----
    v8f acc1[4];
#pragma unroll
    for (int t = 0; t < 4; ++t) acc1[t] = vzero;

    for (int k0 = 0; k0 < 2 * H_DIM; k0 += 4) {
        const int ka = k0 + khalf;
        const v2f a = *(const v2f*)&zbuf[lrow * ZB_LD + ka];
#pragma unroll
        for (int t = 0; t < 4; ++t) {
            const int col = t * 16 + lrow;
            v2f b;
            b.x = fc1_w[ka * 64 + col];
            b.y = fc1_w[(ka + 1) * 64 + col];
            acc1[t] = __builtin_amdgcn_wmma_f32_16x16x4_f32(
                false, a, false, b, (short)0, acc1[t], false, false);
        }
    }
    __syncthreads();                                 // zbuf reads done -> safe to alias z1

#pragma unroll
    for (int t = 0; t < 4; ++t) {
        const int col = t * 16 + lrow;
        const float bias = fc1_b[col];
#pragma unroll
        for (int j = 0; j < 8; ++j)
            z1[(rb + j) * Z1_LD + col] = fmaxf(acc1[t][j] + bias, 0.0f);
    }
    __syncthreads();

    // ---- fc2: [16 x 64] @ [64 x 32] ----
    v8f acc2[2];
#pragma unroll
    for (int t = 0; t < 2; ++t) acc2[t] = vzero;

    for (int k0 = 0; k0 < 64; k0 += 4) {
        const int ka = k0 + khalf;
        const v2f a = *(const v2f*)&z1[lrow * Z1_LD + ka];
#pragma unroll
        for (int t = 0; t < 2; ++t) {
            const int col = t * 16 + lrow;
            v2f b;
            b.x = fc2_w[ka * 32 + col];
            b.y = fc2_w[(ka + 1) * 32 + col];
            acc2[t] = __builtin_amdgcn_wmma_f32_16x16x4_f32(
                false, a, false, b, (short)0, acc2[t], false, false);
        }
    }
    __syncthreads();

#pragma unroll
    for (int t = 0; t < 2; ++t) {
        const int col = t * 16 + lrow;
        const float bias = fc2_b[col];
#pragma unroll
        for (int j = 0; j < 8; ++j)
            z2[(rb + j) * Z2_LD + col] = fmaxf(acc2[t][j] + bias, 0.0f);
    }
    __syncthreads();

    // ---- out: [16 x 32] @ [32 x 1] + sigmoid ----
    if (lane < 16) {
        float s = out_b[0];
#pragma unroll
        for (int c = 0; c < 32; ++c) s += z2[lane * Z2_LD + c] * out_w[c];
        out[row0 + lane] = 1.0f / (1.0f + expf(-s));
    }
}

// --------------------------------------------------------------------------- launch
extern "C" void kernel_launch(void* const* d_in, const int* in_sizes, int n_in,
                              void* d_out, int out_size, void* d_ws, size_t ws_size,
                              hipStream_t stream) {
    const float* x     = (const float*)d_in[0];
    const float* W1    = (const float*)d_in[1];
    const float* b1    = (const float*)d_in[2];
    const float* W2    = (const float*)d_in[3];
    const float* b2    = (const float*)d_in[4];
    const float* fc1_w = (const float*)d_in[5];
    const float* fc1_b = (const float*)d_in[6];
    const float* fc2_w = (const float*)d_in[7];
    const float* fc2_b = (const float*)d_in[8];
    const float* out_w = (const float*)d_in[9];
    const float* out_b = (const float*)d_in[10];
    const int*   ei    = (const int*)d_in[11];     // [2, E]
    const int*   users = (const int*)d_in[12];
    const int*   items = (const int*)d_in[13];

    const int N = in_sizes[0] / D_IN;              // 200000
    const int E = in_sizes[11] / 2;                // 640000
    const int B = in_sizes[12];                    // 65536
    const int* src = ei;
    const int* dst = ei + E;

    float* out = (float*)d_out;

    // workspace layout
    float* deg   = (float*)d_ws;                            // N
    float* dinv  = deg + N;                                 // N
    float* bufA  = dinv + N;                                // N*128
    float* bufB  = bufA + (long long)N * H_DIM;             // N*128
    float* bufC  = bufB + (long long)N * H_DIM;             // N*128
    int*   flags = (int*)(bufC + (long long)N * H_DIM);     // 2048
    int*   numU  = flags + N_ITEMS;                         // 1

    const long long NH  = (long long)N * H_DIM;
    const long long NH4 = NH / 4;
    const long long N4  = (N + 3) / 4;

    // 1. zero deg, agg1 (bufB), item flags
    zero_b128_kernel<<<(int)((N4 + 255) / 256), 256, 0, stream>>>((uint4*)deg, N4);
    zero_b128_kernel<<<(int)((NH4 + 255) / 256), 256, 0, stream>>>((uint4*)bufB, NH4);
    zero_b128_kernel<<<(N_ITEMS / 4 + 255) / 256, 256, 0, stream>>>((uint4*)flags, N_ITEMS / 4);

    // 2. degrees (edges + self-loops), 3. dinv
    degree_kernel<<<(E + N + 255) / 256, 256, 0, stream>>>(dst, deg, E, N);
    dinv_kernel<<<(N + 255) / 256, 256, 0, stream>>>(deg, dinv, N);

    // 4. h_lin1 = x @ W1  (WMMA)
    const int gemm_blocks = (N + 63) / 64;                  // 4 waves/block, 16 rows/wave
    gemm_wmma_kernel<D_IN><<<gemm_blocks, 128, 0, stream>>>(x, W1, bufA, N);

    // 5. agg1[dst] += h_lin1[src]*norm ; 6. h1 = relu(agg1 + b1) -> bufA
    const int edge_blocks = (E + N + 7) / 8;                // 8 edges per 256-thread block
    scatter_kernel<<<edge_blocks, 256, 0, stream>>>(bufA, src, dst, dinv, bufB, E, N);
    bias_act_kernel<true><<<(int)((NH4 + 255) / 256), 256, 0, stream>>>(
        (const float4*)bufB, (const float4*)b1, (float4*)bufA, NH4);

    // 7. re-zero agg (bufB); 8. h_lin2 = h1 @ W2 -> bufC (WMMA)
    zero_b128_kernel<<<(int)((NH4 + 255) / 256), 256, 0, stream>>>((uint4*)bufB, NH4);
    gemm_wmma_kernel<H_DIM><<<gemm_blocks, 128, 0, stream>>>(bufA, W2, bufC, N);

    // 9. agg2[dst] += h_lin2[src]*norm ; 10. h2 = agg2 + b2 -> bufC
    scatter_kernel<<<edge_blocks, 256, 0, stream>>>(bufC, src, dst, dinv, bufB, E, N);
    bias_act_kernel<false><<<(int)((NH4 + 255) / 256), 256, 0, stream>>>(
        (const float4*)bufB, (const float4*)b2, (float4*)bufC, NH4);

    // 11. num_users = N - #unique(items)
    item_flag_kernel<<<(B + 255) / 256, 256, 0, stream>>>(items, flags, B);
    count_unique_kernel<<<1, 256, 0, stream>>>(flags, numU, N, N_ITEMS);

    // 12. MLP scorer (WMMA) -> sigmoid scores
    mlp_wmma_kernel<<<(B + 63) / 64, 128, 0, stream>>>(bufC, users, items, numU,
                                                       fc1_w, fc1_b, fc2_w, fc2_b,
                                                       out_w, out_b, out, B);
}